// GCN_8632884265212
// MI455X (gfx1250) — compile-verified
//
#include <hip/hip_runtime.h>
#include <hip/hip_bf16.h>

#define N_NODES 50000
#define FDIM    128   // aggregation feature dim (all layers aggregate 128-wide)
#define KDIM    128   // GEMM reduction dim (in=hid=128)

typedef __attribute__((ext_vector_type(2))) float v2f;
typedef __attribute__((ext_vector_type(8))) float v8f;

// ---------------------------------------------------------------- utilities
__global__ void zero_f32(float* __restrict__ p, int n) {
    int i = blockIdx.x * blockDim.x + threadIdx.x;
    int stride = gridDim.x * blockDim.x;
    for (; i < n; i += stride) p[i] = 0.0f;
}

// out_deg / in_deg via f32 atomics (L2-resident, 50k-entry tables)
__global__ void degree_kernel(const int* __restrict__ src, const int* __restrict__ dst,
                              float* __restrict__ outdeg, float* __restrict__ indeg, int E) {
    int i = blockIdx.x * blockDim.x + threadIdx.x;
    if (i < E) {
        atomicAdd(&outdeg[src[i]], 1.0f);
        atomicAdd(&indeg[dst[i]], 1.0f);
    }
}

// deg -> clamp(deg,1)^-1/2 in place (norm_src and norm_dst are contiguous)
__global__ void norm_finalize(float* __restrict__ p, int n) {
    int i = blockIdx.x * blockDim.x + threadIdx.x;
    if (i < n) p[i] = rsqrtf(fmaxf(p[i], 1.0f));
}

// ------------------------------------------------------- sparse aggregation
// One wave per edge: lane L holds channels 4L..4L+3 (float4 gather, 4 dword
// atomics). src-degree normalization fused into the gather.
__global__ __launch_bounds__(256)
void aggregate_kernel(const float* __restrict__ x, const int* __restrict__ src,
                      const int* __restrict__ dst, const float* __restrict__ nsrc,
                      float* __restrict__ agg, int E) {
    const int lane = threadIdx.x & 31;
    const int e = blockIdx.x * (blockDim.x >> 5) + (threadIdx.x >> 5);
    if (e >= E) return;                       // wave-uniform
    const int s = src[e];
    const int d = dst[e];
    const float n = nsrc[s];
    const float4 v = ((const float4*)(x + (size_t)s * FDIM))[lane];
    float* o = agg + (size_t)d * FDIM + lane * 4;
    atomicAdd(o + 0, v.x * n);
    atomicAdd(o + 1, v.y * n);
    atomicAdd(o + 2, v.z * n);
    atomicAdd(o + 3, v.w * n);
}

// ------------------------------------------------------------ WMMA GEMM
// out[m, :] = act( norm[m] * (A[m, :] @ W) + bias )
// One wave computes a 16 x NOUT strip using v_wmma_f32_16x16x4_f32.
// A frag (16x4 f32): lanes 0-15 -> rows, K+0/K+1; lanes 16-31 -> rows, K+2/K+3.
// B frag (4x16 f32): VGPR0 = {K+0 | K+2}, VGPR1 = {K+1 | K+3} across half-lanes.
template <int NOUT, bool RELU>
__global__ __launch_bounds__(128)
void gemm_wmma_kernel(const float* __restrict__ A, const float* __restrict__ W,
                      const float* __restrict__ bias, const float* __restrict__ norm,
                      float* __restrict__ out) {
    constexpr int NT = NOUT / 16;
    const int lane  = threadIdx.x & 31;
    const int wave  = threadIdx.x >> 5;
    const int mtile = blockIdx.x * 4 + wave;   // wave-uniform
    if (mtile * 16 >= N_NODES) return;         // wave-uniform guard (EXEC stays full)
    const int m0    = mtile * 16;
    const int r15   = lane & 15;
    const int khalf = lane >> 4;               // 0 or 1
    const int koff  = khalf * 2;               // 0 or 2

    v8f acc[NT];
#pragma unroll
    for (int t = 0; t < NT; ++t)
#pragma unroll
        for (int j = 0; j < 8; ++j) acc[t][j] = 0.0f;

    const float* Ar = A + (size_t)(m0 + r15) * KDIM + koff;
    for (int k = 0; k < KDIM; k += 4) {
        v2f a;
        a.x = Ar[k];
        a.y = Ar[k + 1];
#pragma unroll
        for (int t = 0; t < NT; ++t) {
            const float* Wp = W + (size_t)(k + koff) * NOUT + t * 16 + r15;
            v2f b;
            b.x = Wp[0];
            b.y = Wp[NOUT];
            acc[t] = __builtin_amdgcn_wmma_f32_16x16x4_f32(
                false, a, false, b, (short)0, acc[t], false, false);
        }
    }

    // epilogue: row scale by norm_dst, + bias, optional ReLU
    const int rbase = m0 + khalf * 8;          // C/D: lanes 16-31 hold rows M+8..M+15
    float nrm[8];
#pragma unroll
    for (int j = 0; j < 8; ++j) nrm[j] = norm[rbase + j];
#pragma unroll
    for (int t = 0; t < NT; ++t) {
        const float bv = bias[t * 16 + r15];
#pragma unroll
        for (int j = 0; j < 8; ++j) {
            float v = acc[t][j] * nrm[j] + bv;
            if (RELU) v = fmaxf(v, 0.0f);
            out[(size_t)(rbase + j) * NOUT + t * 16 + r15] = v;
        }
    }
}

// ---------------------------------------------------------------- launcher
extern "C" void kernel_launch(void* const* d_in, const int* in_sizes, int n_in,
                              void* d_out, int out_size, void* d_ws, size_t ws_size,
                              hipStream_t stream) {
    const float* features = (const float*)d_in[0];
    const float* W1 = (const float*)d_in[1];
    const float* b1 = (const float*)d_in[2];
    const float* W2 = (const float*)d_in[3];
    const float* b2 = (const float*)d_in[4];
    const float* W3 = (const float*)d_in[5];
    const float* b3 = (const float*)d_in[6];
    const int*  src = (const int*)d_in[7];
    const int*  dst = (const int*)d_in[8];
    const int E = in_sizes[7];

    float* ws       = (float*)d_ws;
    float* norm_src = ws;                                 // [N]
    float* norm_dst = ws + N_NODES;                       // [N]
    float* aggA     = ws + 2 * N_NODES;                   // [N,128]
    float* bufB     = aggA + (size_t)N_NODES * FDIM;      // [N,128]

    // degrees + symmetric norms (shared by all 3 layers)
    zero_f32<<<512, 256, 0, stream>>>(norm_src, 2 * N_NODES);
    degree_kernel<<<(E + 255) / 256, 256, 0, stream>>>(src, dst, norm_src, norm_dst, E);
    norm_finalize<<<(2 * N_NODES + 255) / 256, 256, 0, stream>>>(norm_src, 2 * N_NODES);

    const int aggBlocks  = (E + 7) / 8;                   // 8 waves (edges) per block
    const int gemmBlocks = (N_NODES / 16 + 3) / 4;        // 4 M-tiles per block

    // layer 1: features -> bufB (relu)
    zero_f32<<<2048, 256, 0, stream>>>(aggA, N_NODES * FDIM);
    aggregate_kernel<<<aggBlocks, 256, 0, stream>>>(features, src, dst, norm_src, aggA, E);
    gemm_wmma_kernel<128, true><<<gemmBlocks, 128, 0, stream>>>(aggA, W1, b1, norm_dst, bufB);

    // layer 2: bufB -> bufB (relu)
    zero_f32<<<2048, 256, 0, stream>>>(aggA, N_NODES * FDIM);
    aggregate_kernel<<<aggBlocks, 256, 0, stream>>>(bufB, src, dst, norm_src, aggA, E);
    gemm_wmma_kernel<128, true><<<gemmBlocks, 128, 0, stream>>>(aggA, W2, b2, norm_dst, bufB);

    // layer 3: bufB -> d_out (no relu, NOUT=64)
    zero_f32<<<2048, 256, 0, stream>>>(aggA, N_NODES * FDIM);
    aggregate_kernel<<<aggBlocks, 256, 0, stream>>>(bufB, src, dst, norm_src, aggA, E);
    gemm_wmma_kernel<64, false><<<gemmBlocks, 128, 0, stream>>>(aggA, W3, b3, norm_dst,
                                                                (float*)d_out);
}